// AdapGConv_81131932221716
// MI455X (gfx1250) — compile-verified
//
#include <hip/hip_runtime.h>

#define D_IN   512
#define D_OUT  512
#define NNODES 10000
#define NEDGES 160000

// register-blocking of the WMMA GEMM: 5 M-tiles x 2 N-tiles per wave
#define MT 5
#define NT 2

typedef __attribute__((ext_vector_type(2))) float v2f;
typedef __attribute__((ext_vector_type(8))) float v8f;

// ---------------------------------------------------------------------------
// Kernel 0: zero the f32 accumulation buffer (agg) in workspace.
// ---------------------------------------------------------------------------
__global__ void zero_f32x4_kernel(float4* __restrict__ p, int n4) {
    int i = blockIdx.x * blockDim.x + threadIdx.x;
    if (i < n4) p[i] = make_float4(0.f, 0.f, 0.f, 0.f);
}

// ---------------------------------------------------------------------------
// Kernel 1: fused importance-sampling scale + edge scatter (segment_sum).
// One wave (32 lanes) per edge. Per-edge scalar s = val[e] / (q[col]*N).
// Source row read as coalesced float4s (feat = 20.5MB, L2-resident in 192MB),
// destination accumulated with global_atomic_add_f32.
// ---------------------------------------------------------------------------
__global__ void scatter_kernel(const float* __restrict__ feat,
                               const float* __restrict__ q,
                               const float* __restrict__ val,
                               const int*   __restrict__ erow,
                               const int*   __restrict__ ecol,
                               const int*   __restrict__ nnum,
                               float* __restrict__ agg) {
    int wave = (blockIdx.x * blockDim.x + threadIdx.x) >> 5;
    int lane = threadIdx.x & 31;
    if (wave >= NEDGES) return;

    int r = erow[wave];
    int c = ecol[wave];
    float s = val[wave] / (q[c] * (float)(*nnum));

    const float4* src = (const float4*)(feat + (size_t)c * D_IN);
    float*        dst = agg + (size_t)r * D_IN;

#pragma unroll
    for (int j = 0; j < 4; ++j) {
        int idx = j * 32 + lane;          // float4 slot in [0,128)
        float4 v = src[idx];
        atomicAdd(dst + idx * 4 + 0, v.x * s);
        atomicAdd(dst + idx * 4 + 1, v.y * s);
        atomicAdd(dst + idx * 4 + 2, v.z * s);
        atomicAdd(dst + idx * 4 + 3, v.w * s);
    }
}

// ---------------------------------------------------------------------------
// Kernel 2: out = relu(agg @ W + b) in full f32 via V_WMMA_F32_16X16X4_F32.
// Register-blocked: each wave owns MT x NT = 5x2 16x16 output tiles, so each
// K=4 step issues 10 wmma against 5 A-frag b64 loads + 4 B b32 loads.
// 125 M-groups x 16 N-groups = 2000 waves exactly (EXEC all-1s everywhere).
//
// Operand layouts per CDNA5 ISA 7.12.2 (f32, 16x4 A / 4x16 B / 16x16 C):
//   A: lanes 0-15 -> M=lane, VGPR0/1 = K=0/1; lanes 16-31 -> K=2/3
//   B: lanes 0-15 -> N=lane, VGPR0/1 = K=0/1; lanes 16-31 -> K=2/3
//   C: VGPR i -> M = i + 8*(lane>=16), N = lane&15
// ---------------------------------------------------------------------------
__global__ void __launch_bounds__(256)
wmma_gemm_bias_relu_kernel(const float* __restrict__ A,     // agg  [NNODES, D_IN]
                           const float* __restrict__ W,     // [D_IN, D_OUT] row-major
                           const float* __restrict__ bias,  // [D_OUT]
                           float* __restrict__ out) {       // [NNODES, D_OUT]
    const int nGroups = (D_OUT / 16) / NT;                   // 16
    int wid  = (blockIdx.x * blockDim.x + threadIdx.x) >> 5; // global wave id
    int lane = threadIdx.x & 31;

    int mg = wid / nGroups;                 // 0..124  (uniform per wave)
    if (mg >= (NNODES / 16) / MT) return;   // never taken with exact grid
    int ng = wid % nGroups;

    int m0 = mg * (MT * 16);                // base row of the 80-row M panel
    int tn = ng * (NT * 16);                // base col of the 32-col N panel

    int half = lane >> 4;                   // 0: K=0,1   1: K=2,3
    int l15  = lane & 15;

    // Single moving base pointers; per-tile deltas (t*16*D_IN*4 = 32KB for A,
    // u*16*4 = 64B and +D_OUT*4 = 2KB for B) fold into 24-bit inst offsets.
    const float* abase = A + (size_t)(m0 + l15) * D_IN + 2 * half;
    const float* wbase = W + (size_t)(2 * half) * D_OUT + tn + l15;

    v8f acc[MT][NT];
#pragma unroll
    for (int t = 0; t < MT; ++t)
#pragma unroll
        for (int u = 0; u < NT; ++u)
            acc[t][u] = (v8f){};

#pragma unroll 2
    for (int k = 0; k < D_IN; k += 4) {
        v2f a[MT];
#pragma unroll
        for (int t = 0; t < MT; ++t)
            a[t] = *(const v2f*)(abase + (size_t)t * 16 * D_IN + k);

        v2f b[NT];
#pragma unroll
        for (int u = 0; u < NT; ++u) {
            const float* wp = wbase + (size_t)k * D_OUT + u * 16;
            b[u].x = wp[0];        // W[k+2h  , n]
            b[u].y = wp[D_OUT];    // W[k+2h+1, n]
        }

#pragma unroll
        for (int t = 0; t < MT; ++t)
#pragma unroll
            for (int u = 0; u < NT; ++u)
                acc[t][u] = __builtin_amdgcn_wmma_f32_16x16x4_f32(
                    /*neg_a=*/false, a[t], /*neg_b=*/false, b[u],
                    /*c_mod=*/(short)0, acc[t][u],
                    /*reuse_a=*/false, /*reuse_b=*/false);
    }

    // Epilogue: bias + relu, lane-striped f32 stores per C layout.
    float bv[NT];
#pragma unroll
    for (int u = 0; u < NT; ++u)
        bv[u] = bias[tn + u * 16 + l15];

#pragma unroll
    for (int t = 0; t < MT; ++t) {
#pragma unroll
        for (int u = 0; u < NT; ++u) {
#pragma unroll
            for (int i = 0; i < 8; ++i) {
                int row = m0 + t * 16 + i + 8 * half;
                int col = tn + u * 16 + l15;
                float v = acc[t][u][i] + bv[u];
                out[(size_t)row * D_OUT + col] = v > 0.f ? v : 0.f;
            }
        }
    }
}

// ---------------------------------------------------------------------------
// Host launcher
// ---------------------------------------------------------------------------
extern "C" void kernel_launch(void* const* d_in, const int* in_sizes, int n_in,
                              void* d_out, int out_size, void* d_ws, size_t ws_size,
                              hipStream_t stream) {
    const float* feat = (const float*)d_in[0];   // [NNODES, D_IN]
    const float* q    = (const float*)d_in[1];   // [NNODES]
    const float* val  = (const float*)d_in[2];   // [NEDGES]
    const float* W    = (const float*)d_in[3];   // [D_IN, D_OUT]
    const float* bias = (const float*)d_in[4];   // [D_OUT]
    const int*   erow = (const int*)d_in[5];     // [NEDGES]
    const int*   ecol = (const int*)d_in[6];     // [NEDGES]
    const int*   nnum = (const int*)d_in[7];     // scalar

    float* out = (float*)d_out;
    float* agg = (float*)d_ws;                   // [NNODES, D_IN] f32 = 20.48 MB

    // 1) zero accumulator
    int n4 = NNODES * D_IN / 4;
    zero_f32x4_kernel<<<(n4 + 255) / 256, 256, 0, stream>>>((float4*)agg, n4);

    // 2) fused scale + scatter: one wave per edge, 8 waves per block
    int scatter_blocks = (NEDGES * 32 + 255) / 256;        // 20000
    scatter_kernel<<<scatter_blocks, 256, 0, stream>>>(feat, q, val, erow, ecol,
                                                       nnum, agg);

    // 3) WMMA f32 GEMM + bias + relu: 125 * 16 = 2000 waves -> 250 blocks
    int gemm_waves  = ((NNODES / 16) / MT) * ((D_OUT / 16) / NT);  // 2000
    int gemm_blocks = gemm_waves / 8;                              // 250
    wmma_gemm_bias_relu_kernel<<<gemm_blocks, 256, 0, stream>>>(agg, W, bias, out);
}